// FRQI_Encoding_11940009083329
// MI455X (gfx1250) — compile-verified
//
#include <hip/hip_runtime.h>

// cos(2*theta) with theta = p * (pi/2/255)  ==>  cos(p * pi/255)
#define PI_OVER_255 0.01231997132f

typedef float vf2 __attribute__((ext_vector_type(2)));
typedef float v8f __attribute__((ext_vector_type(8)));

// FRQI closed form:
//   out[b,0] = out[b,1] = 0  (address-qubit <Z> are input-independent and equal 0 for 2 addr qubits)
//   out[b,2] = mean_i cos(pix[b,i] * pi/255)
//
// Memory-bound streaming kernel: 64 MiB in + 48 MiB out @ 23.3 TB/s ~ 5 us floor.
// Each wave32 handles 32 consecutive batches:
//   group g in {0,1}: lane L supplies pixel-pair (L>>4) of batch base+g*16+(L&15)
//   -> cosines land in the two B-matrix VGPRs of V_WMMA_F32_16X16X4_F32.
//   A = 0.25 everywhere (layout-independent), C = 0  =>  D[m,n] = mean of column n,
//   replicated over m, so D[0] per lane = mean(batch base+g*16+(L%16)).
__global__ __launch_bounds__(256) void frqi_encoding_kernel(
    const float* __restrict__ in, float* __restrict__ out, int n_batch) {
  const int tid  = threadIdx.x;
  const int lane = tid & 31;
  const int wbase = (blockIdx.x * 8 + (tid >> 5)) << 5;  // first batch of this wave
  if (wbase >= n_batch) return;                           // wave-uniform guard (EXEC stays full)

  const int n0   = wbase + (lane & 15);   // group-0 batch owned by this lane's column
  const int n1   = n0 + 16;               // group-1 batch
  const int half = lane >> 4;             // 0 -> pixels {0,1}, 1 -> pixels {2,3}

  // Coalesced 64-bit NT loads: each instruction covers 256 contiguous bytes per wave.
  vf2 q0 = __builtin_nontemporal_load((const vf2*)(in + n0 * 4 + half * 2));
  vf2 q1 = __builtin_nontemporal_load((const vf2*)(in + n1 * 4 + half * 2));

  vf2 b0, b1;  // B-matrix operands (cosines), column n = lanes {n, n+16}
  b0.x = __cosf(q0.x * PI_OVER_255);
  b0.y = __cosf(q0.y * PI_OVER_255);
  b1.x = __cosf(q1.x * PI_OVER_255);
  b1.y = __cosf(q1.y * PI_OVER_255);

  float m;
#if __has_builtin(__builtin_amdgcn_wmma_f32_16x16x4_f32)
  // Exact f32 4-tap mean on the matrix pipe: D = A(=0.25) x B(cosines) + 0.
  vf2 a = {0.25f, 0.25f};
  v8f c = {};
  v8f d0 = __builtin_amdgcn_wmma_f32_16x16x4_f32(
      /*neg_a=*/false, a, /*neg_b=*/false, b0,
      /*c_mod=*/(short)0, c, /*reuse_a=*/false, /*reuse_b=*/false);
  v8f d1 = __builtin_amdgcn_wmma_f32_16x16x4_f32(
      /*neg_a=*/false, a, /*neg_b=*/false, b1,
      /*c_mod=*/(short)0, c, /*reuse_a=*/false, /*reuse_b=*/false);
  // d0[0] = mean(batch wbase + (lane%16)); d1[0] = mean(batch wbase+16 + (lane%16)).
  m = (lane < 16) ? d0[0] : d1[0];
#else
  // VALU fallback: butterfly across half-waves (wave32).
  float s0 = b0.x + b0.y;
  float s1 = b1.x + b1.y;
  float t0 = 0.25f * (s0 + __shfl_xor(s0, 16));
  float t1 = 0.25f * (s1 + __shfl_xor(s1, 16));
  m = (lane < 16) ? t0 : t1;
#endif

  // Lane L owns batch wbase+L: store {0, 0, mean} -> 384 contiguous bytes per wave.
  float* op = out + (size_t)(wbase + lane) * 3;
  __builtin_nontemporal_store(0.0f, op + 0);
  __builtin_nontemporal_store(0.0f, op + 1);
  __builtin_nontemporal_store(m,    op + 2);
}

extern "C" void kernel_launch(void* const* d_in, const int* in_sizes, int n_in,
                              void* d_out, int out_size, void* d_ws, size_t ws_size,
                              hipStream_t stream) {
  const float* in = (const float*)d_in[0];
  float* out = (float*)d_out;
  const int n_batch = in_sizes[0] / 4;              // (B, 2, 2) floats
  const int blocks  = (n_batch + 255) / 256;        // 256 thr = 8 wave32 = 256 batches/block
  frqi_encoding_kernel<<<blocks, 256, 0, stream>>>(in, out, n_batch);
}